// TorchFeatureTransformer_21294447854185
// MI455X (gfx1250) — compile-verified
//
#include <hip/hip_runtime.h>
#include <stdint.h>

#define BATCH   32768
#define NNODES  16
#define LATENT  8
#define DCOL    128      // 16*8
#define NPAIR   120      // triu(16,1)
#define NFEAT   3872     // 128 + 14*120 + 5*128 + 16 + 448 + 960
#define NPOLY   1000
#define NSEL    512
#define NPART   128      // stat blocks
#define ROWS_PER_STAT (BATCH / NPART)   // 256
#define ROWS_PER_EMIT 16
#define SOFTC   1.0e-3f
#define NGRP    (NFEAT / 16)            // 242

typedef __attribute__((ext_vector_type(2))) float v2f;
typedef __attribute__((ext_vector_type(8))) float v8f;
typedef __attribute__((ext_vector_type(4))) int   v4i;

#define AS_GLOBAL __attribute__((address_space(1)))
#define AS_LDS    __attribute__((address_space(3)))

#if defined(__gfx1250__) && __has_builtin(__builtin_amdgcn_wmma_f32_16x16x4_f32)
#define HAVE_WMMA4 1
#else
#define HAVE_WMMA4 0
#endif

#if defined(__gfx1250__) && __has_builtin(__builtin_amdgcn_global_load_async_to_lds_b128)
#define HAVE_ASYNC 1
#else
#define HAVE_ASYNC 0
#endif

__device__ __forceinline__ void stage16B(const float* g, float* l) {
#if HAVE_ASYNC
  __builtin_amdgcn_global_load_async_to_lds_b128(
      (AS_GLOBAL v4i*)(void*)(g),
      (AS_LDS    v4i*)(void*)(l), 0, 0);
#else
  l[0] = g[0]; l[1] = g[1]; l[2] = g[2]; l[3] = g[3];
#endif
}

__device__ __forceinline__ void wait_async_done() {
#if HAVE_ASYNC
#if __has_builtin(__builtin_amdgcn_s_wait_asynccnt)
  __builtin_amdgcn_s_wait_asynccnt(0);
#else
  asm volatile("s_wait_asynccnt 0" ::: "memory");
#endif
#endif
}

__device__ __forceinline__ float clip1e6(float v) {
  return fminf(fmaxf(v, -1.0e6f), 1.0e6f);
}

__device__ __forceinline__ float ipowf(float b, int n) {
  float r = 1.0f;
  while (n > 0) { if (n & 1) r *= b; b *= b; n >>= 1; }
  return r;
}

__device__ __forceinline__ void pair16(int p, int* i, int* j) {
  int a = 0, rem = p;
  while (rem >= 15 - a) { rem -= 15 - a; ++a; }
  *i = a; *j = a + 1 + rem;
}

// nan_to_num(nan=0, posinf=1e9, neginf=-1e9) then clip +-1e12
__device__ __forceinline__ float finalize_feat(float v) {
  if (!(v == v)) v = 0.0f;
  else if (isinf(v)) v = (v > 0.0f) ? 1.0e9f : -1.0e9f;
  return fminf(fmaxf(v, -1.0e12f), 1.0e12f);
}

// Column decode, exactly in the reference concatenation order.
// zn = normalized (from clipped z); dd = pair distances (already +1e-6).
__device__ float eval_feature(int c, const float* zn, const float* dd) {
  if (c < DCOL) return clip1e6(zn[c]);                 // X: zn part (clipped)
  c -= DCOL;
  if (c < 14 * NPAIR) {                                // X: dist families (clipped)
    int g = c / NPAIR, p = c - g * NPAIR;
    float d = dd[p];
    float v;
    if (g == 0)       v = d;
    else if (g == 1)  v = 1.0f / (d + SOFTC);
    else if (g <= 10) {
      const int pw[9] = {2, 3, 4, 5, 6, 8, 10, 12, 14};
      v = 1.0f / (ipowf(d, pw[g - 2]) + SOFTC);
    }
    else if (g == 11) v = expf(-d);
    else if (g == 12) v = expf(-d) / (d + SOFTC);
    else              v = logf(d + SOFTC);
    return clip1e6(v);
  }
  c -= 14 * NPAIR;
  if (c < DCOL) { float t = zn[c]; return t * t; }     // zn^2
  c -= DCOL;
  if (c < DCOL) return sinf(zn[c]);                    // sin
  c -= DCOL;
  if (c < DCOL) return cosf(zn[c]);                    // cos
  c -= DCOL;
  if (c < DCOL) return logf(fabsf(zn[c]) + 0.001f);    // log|zn|+1e-3
  c -= DCOL;
  if (c < DCOL) {                                      // exp(clip(zn,-10,2))
    float t = fminf(fmaxf(zn[c], -10.0f), 2.0f);
    return expf(t);
  }
  c -= DCOL;
  if (c < NNODES) {                                    // p_sq
    const float* x = zn + c * LATENT;
    return x[4]*x[4] + x[5]*x[5] + x[6]*x[6] + x[7]*x[7];
  }
  c -= NNODES;
  if (c < NNODES * 28) {                               // intra: node*28 + q
    int node = c / 28, q = c - node * 28;
    int a = 0, rem = q;
    while (rem >= 7 - a) { rem -= 7 - a; ++a; }
    int b = a + 1 + rem;
    const float* x = zn + node * LATENT;
    return x[a] * x[b];
  }
  c -= NNODES * 28;                                    // inter: pair*8 + dim
  int p = c >> 3, dim = c & 7;
  int i, j; pair16(p, &i, &j);
  return zn[i * LATENT + dim] * zn[j * LATENT + dim];
}

// ---------------- Phase A: per-block column sum / sumsq partials ----------------
__global__ __launch_bounds__(256)
void stats_kernel(const float* __restrict__ z, const float* __restrict__ zmean,
                  const float* __restrict__ zstd, float* __restrict__ psum,
                  float* __restrict__ psq) {
  __shared__ float accS[NFEAT];
  __shared__ float accQ[NFEAT];
  __shared__ float Zr[DCOL];
  __shared__ float ZN[DCOL];
  __shared__ float Dd[NPAIR];
  __shared__ float zmS[DCOL];
  __shared__ float zsS[DCOL];
  const int tid = threadIdx.x;
  for (int c = tid; c < NFEAT; c += 256) { accS[c] = 0.0f; accQ[c] = 0.0f; }
  if (tid < DCOL) { zmS[tid] = zmean[tid]; zsS[tid] = zstd[tid]; }
  __syncthreads();
  const int row0 = blockIdx.x * ROWS_PER_STAT;
  for (int rr = 0; rr < ROWS_PER_STAT; ++rr) {
    const float* zrow = z + (size_t)(row0 + rr) * DCOL;
    if (tid < 32) {                       // wave 0 stages the 512B row into LDS
      stage16B(zrow + tid * 4, Zr + tid * 4);
      wait_async_done();
    }
    __syncthreads();
    if (tid < DCOL) {
      float v = clip1e6(Zr[tid]);
      ZN[tid] = (v - zmS[tid]) / zsS[tid];
    } else if (tid < DCOL + NPAIR) {
      int p = tid - DCOL;
      int i, j; pair16(p, &i, &j);
      float dx = clip1e6(Zr[i*LATENT+0]) - clip1e6(Zr[j*LATENT+0]);
      float dy = clip1e6(Zr[i*LATENT+1]) - clip1e6(Zr[j*LATENT+1]);
      dx -= 10.0f * rintf(dx / 10.0f);    // minimum image, half-to-even like jnp.round
      dy -= 10.0f * rintf(dy / 10.0f);
      Dd[p] = sqrtf(dx*dx + dy*dy) + 1.0e-6f;
    }
    __syncthreads();
    for (int c = tid; c < NFEAT; c += 256) {   // each column owned by one thread
      float v = eval_feature(c, ZN, Dd);
      accS[c] += v;
      accQ[c] += v * v;
    }
    __syncthreads();
  }
  float* ps = psum + (size_t)blockIdx.x * NFEAT;
  float* pq = psq  + (size_t)blockIdx.x * NFEAT;
  for (int c = tid; c < NFEAT; c += 256) { ps[c] = accS[c]; pq[c] = accQ[c]; }
}

// ------- Phase B: WMMA ones-reduction over 128 partials, variance, rank, select -------
__global__ __launch_bounds__(256)
void select_kernel(const float* __restrict__ psum, const float* __restrict__ psq,
                   const int* __restrict__ fmask, const float* __restrict__ xpm,
                   const float* __restrict__ xps, int* __restrict__ selcol,
                   float* __restrict__ selm, float* __restrict__ sels) {
  __shared__ float tsum[NFEAT];
  __shared__ float tsq[NFEAT];
  __shared__ float varr[NFEAT];
  __shared__ int   topi[NPOLY];
  const int tid = threadIdx.x;
#if HAVE_WMMA4
  // D = ones(16x4) x B(4x16) + C  =>  every D row = column sums of B.
  const int lane = tid & 31;
  const int wave = tid >> 5;
  const int half = lane >> 4;
  const int n    = lane & 15;
  v2f ones; ones.x = 1.0f; ones.y = 1.0f;
  for (int g = wave; g < NGRP; g += 8) {       // wave-uniform trip counts
    const int col = g * 16 + n;
    v8f aS = {0,0,0,0,0,0,0,0};
    v8f aQ = {0,0,0,0,0,0,0,0};
    for (int pb = 0; pb < NPART; pb += 4) {
      const int r0 = pb + 2 * half;            // K rows covered per half-wave
      v2f bS, bQ;
      bS.x = psum[(size_t)r0 * NFEAT + col];
      bS.y = psum[(size_t)(r0 + 1) * NFEAT + col];
      bQ.x = psq [(size_t)r0 * NFEAT + col];
      bQ.y = psq [(size_t)(r0 + 1) * NFEAT + col];
      aS = __builtin_amdgcn_wmma_f32_16x16x4_f32(false, ones, false, bS, (short)0, aS, false, false);
      aQ = __builtin_amdgcn_wmma_f32_16x16x4_f32(false, ones, false, bQ, (short)0, aQ, false, false);
    }
    if (lane < 16) { tsum[col] = aS[0]; tsq[col] = aQ[0]; }
  }
#else
  for (int c = tid; c < NFEAT; c += 256) {
    float s = 0.0f, q = 0.0f;
    for (int p = 0; p < NPART; ++p) { s += psum[(size_t)p*NFEAT+c]; q += psq[(size_t)p*NFEAT+c]; }
    tsum[c] = s; tsq[c] = q;
  }
#endif
  __syncthreads();
  for (int c = tid; c < NFEAT; c += 256) {     // var (ddof=1), finalized in f64
    double s = (double)tsum[c];
    double q = (double)tsq[c];
    double mean = s / (double)BATCH;
    varr[c] = (float)((q - s * mean) / (double)(BATCH - 1));
  }
  __syncthreads();
  // Exact top_k semantics: rank = #(vj > vc) + #(vj == vc && j < c); deterministic.
  for (int c = tid; c < NFEAT; c += 256) {
    const float vc = varr[c];
    int rank = 0;
    for (int j = 0; j < NFEAT; ++j) {
      float vj = varr[j];
      rank += (vj > vc) ? 1 : ((vj == vc && j < c) ? 1 : 0);
    }
    if (rank < NPOLY) topi[rank] = c;
  }
  __syncthreads();
  for (int j = tid; j < NSEL; j += 256) {
    int fm = fmask[j];                          // index into the top-1000 ordering
    selcol[j] = topi[fm];
    selm[j] = xpm[fm];
    sels[j] = xps[fm];
  }
}

// ---------------- Phase C: compute only the 512 selected columns per row ----------------
__global__ __launch_bounds__(256)
void emit_kernel(const float* __restrict__ z, const float* __restrict__ zmean,
                 const float* __restrict__ zstd, const int* __restrict__ selcol,
                 const float* __restrict__ selm, const float* __restrict__ sels,
                 float* __restrict__ out) {
  __shared__ float ZB[ROWS_PER_EMIT * DCOL];   // raw z, then zn in place
  __shared__ float DB[ROWS_PER_EMIT * NPAIR];
  __shared__ int   sc[NSEL];
  __shared__ float sm[NSEL];
  __shared__ float ss[NSEL];
  __shared__ float zmS[DCOL];
  __shared__ float zsS[DCOL];
  const int tid = threadIdx.x;
  for (int j = tid; j < NSEL; j += 256) { sc[j] = selcol[j]; sm[j] = selm[j]; ss[j] = sels[j]; }
  if (tid < DCOL) { zmS[tid] = zmean[tid]; zsS[tid] = zstd[tid]; }
  const float* zr = z + (size_t)blockIdx.x * ROWS_PER_EMIT * DCOL;
  for (int ch = tid; ch < (ROWS_PER_EMIT * DCOL) / 4; ch += 256) {
    stage16B(zr + ch * 4, ZB + ch * 4);        // async 16B chunks -> LDS
  }
  wait_async_done();
  __syncthreads();
  for (int t = tid; t < ROWS_PER_EMIT * NPAIR; t += 256) {
    int r = t / NPAIR, p = t - r * NPAIR;
    int i, j; pair16(p, &i, &j);
    const float* zz = ZB + r * DCOL;
    float dx = clip1e6(zz[i*LATENT+0]) - clip1e6(zz[j*LATENT+0]);
    float dy = clip1e6(zz[i*LATENT+1]) - clip1e6(zz[j*LATENT+1]);
    dx -= 10.0f * rintf(dx / 10.0f);
    dy -= 10.0f * rintf(dy / 10.0f);
    DB[t] = sqrtf(dx*dx + dy*dy) + 1.0e-6f;
  }
  __syncthreads();
  for (int t = tid; t < ROWS_PER_EMIT * DCOL; t += 256) {  // raw -> zn in place
    int k = t & (DCOL - 1);
    float v = clip1e6(ZB[t]);
    ZB[t] = (v - zmS[k]) / zsS[k];
  }
  __syncthreads();
  float* orow = out + (size_t)blockIdx.x * ROWS_PER_EMIT * NSEL;
  for (int t = tid; t < ROWS_PER_EMIT * NSEL; t += 256) {
    int r = t >> 9, j = t & (NSEL - 1);
    float v = eval_feature(sc[j], ZB + r * DCOL, DB + r * NPAIR);
    v = finalize_feat(v);
    orow[t] = (v - sm[j]) / ss[j];
  }
}

extern "C" void kernel_launch(void* const* d_in, const int* in_sizes, int n_in,
                              void* d_out, int out_size, void* d_ws, size_t ws_size,
                              hipStream_t stream) {
  (void)in_sizes; (void)n_in; (void)out_size; (void)ws_size;
  const float* z     = (const float*)d_in[0];
  const float* zmean = (const float*)d_in[1];
  const float* zstd  = (const float*)d_in[2];
  const float* xpm   = (const float*)d_in[3];
  const float* xps   = (const float*)d_in[4];
  const int*   fmask = (const int*)d_in[5];
  float* out  = (float*)d_out;
  // workspace layout (~4 MB): psum[128][3872] | psq[128][3872] | selcol | selm | sels
  float* psum = (float*)d_ws;
  float* psq  = psum + (size_t)NPART * NFEAT;
  int*   selc = (int*)(psq + (size_t)NPART * NFEAT);
  float* selm = (float*)(selc + NSEL);
  float* sels = selm + NSEL;
  stats_kernel <<<NPART, 256, 0, stream>>>(z, zmean, zstd, psum, psq);
  select_kernel<<<1, 256, 0, stream>>>(psum, psq, fmask, xpm, xps, selc, selm, sels);
  emit_kernel  <<<BATCH / ROWS_PER_EMIT, 256, 0, stream>>>(z, zmean, zstd, selc, selm, sels, out);
}